// SSIM_6167573037195
// MI455X (gfx1250) — compile-verified
//
#include <hip/hip_runtime.h>

// CDNA5 / gfx1250 SSIM (7x7 uniform filter, VALID) via fp32 WMMA.
// Separable box filter expressed as GEMM against constant banded (1/7) matrices
// using V_WMMA_F32_16X16X4_F32 so everything stays fp32 (variance terms are
// cancellation-sensitive). One wave per 16x16 output tile, wave-private LDS.

typedef float v2f __attribute__((ext_vector_type(2)));
typedef float v8f __attribute__((ext_vector_type(8)));

#define WAVES      4          // waves (tiles) per block
#define TS         26         // x/y LDS tile row stride (floats), 8B-aligned v2f reads
#define XT_FLOATS  (24 * TS)  // 24 rows (22 valid + 2 zero K-pad rows)
#define HS         26         // transposed-H stride per column
#define HT_FLOATS  (16 * HS)  // 16 cols x (24 rows used)
#define WAVE_LDS   (2 * XT_FLOATS + 5 * HT_FLOATS)   // 3328 floats / wave

__device__ __forceinline__ v8f vzero8() {
  v8f z = {0.f, 0.f, 0.f, 0.f, 0.f, 0.f, 0.f, 0.f};
  return z;
}

__global__ __launch_bounds__(32 * WAVES)
void ssim_wmma_kernel(const float* __restrict__ X, const float* __restrict__ Y,
                      float* __restrict__ out) {
  __shared__ float lds[WAVES][WAVE_LDS];

  const int lane = threadIdx.x & 31;
  const int wave = threadIdx.x >> 5;
  const int hi   = lane >> 4;    // 0: lanes 0-15, 1: lanes 16-31
  const int lo   = lane & 15;

  float* xts = &lds[wave][0];
  float* yts = xts + XT_FLOATS;
  float* hts = yts + XT_FLOATS;

  // tile decomposition: 96 images x 32 x 32 tiles
  const int t   = blockIdx.x * WAVES + wave;
  const int img = t >> 10;
  const int rem = t & 1023;
  const int r0  = (rem >> 5) << 4;
  const int c0  = (rem & 31) << 4;

  const float* Xb = X + img * (512 * 512);
  const float* Yb = Y + img * (512 * 512);

  // Load 22x22 halo (clamped addresses; zero the K-pad rows/cols 22..23).
  for (int idx = lane; idx < 24 * 24; idx += 32) {
    const int r = idx / 24;
    const int c = idx - r * 24;
    const int gr = min(r0 + r, 511);
    const int gc = min(c0 + c, 511);
    const bool inb = (r < 22) && (c < 22);
    const int g = gr * 512 + gc;
    xts[r * TS + c] = inb ? Xb[g] : 0.0f;
    yts[r * TS + c] = inb ? Yb[g] : 0.0f;
  }
  // Wave-private LDS regions; in-wave DS ordering suffices (no barrier).

  // Constant banded filter fragments. w(j,c) = (c <= j <= c+6) ? 1/7 : 0.
  // Serves as B (horizontal, Wh) and as A (vertical, Wv = Wh^T) — same lane values.
  v2f wB[6];
#pragma unroll
  for (int k = 0; k < 6; ++k) {
    const int j0 = 4 * k + 2 * hi;
    const float b0 = (j0 >= lo && j0 <= lo + 6) ? (1.0f / 7.0f) : 0.0f;
    const float b1 = (j0 + 1 >= lo && j0 + 1 <= lo + 6) ? (1.0f / 7.0f) : 0.0f;
    v2f w = {b0, b1};
    wB[k] = w;
  }

  // ---- Horizontal pass: H(22x16) = M(22x24) x Wh(24x16), two 16-row M-tiles.
  v8f h0[5], h1[5];
#pragma unroll
  for (int m = 0; m < 5; ++m) { h0[m] = vzero8(); h1[m] = vzero8(); }

#pragma unroll
  for (int k = 0; k < 6; ++k) {
    const int col = 4 * k + 2 * hi;        // A-fragment K columns (col, col+1)
    {
      const int row = lo;                  // M-tile 0: rows 0..15
      const v2f xa = *(const v2f*)(xts + row * TS + col);
      const v2f ya = *(const v2f*)(yts + row * TS + col);
      v2f A[5] = { xa, ya, xa * xa, ya * ya, xa * ya };
#pragma unroll
      for (int m = 0; m < 5; ++m)
        h0[m] = __builtin_amdgcn_wmma_f32_16x16x4_f32(
            false, A[m], false, wB[k], (short)0, h0[m], false, false);
    }
    {
      const int row = 16 + lo;             // M-tile 1: rows 16..31 (22/23 zero-padded)
      const v2f xa = *(const v2f*)(xts + row * TS + col);
      const v2f ya = *(const v2f*)(yts + row * TS + col);
      v2f A[5] = { xa, ya, xa * xa, ya * ya, xa * ya };
#pragma unroll
      for (int m = 0; m < 5; ++m)
        h1[m] = __builtin_amdgcn_wmma_f32_16x16x4_f32(
            false, A[m], false, wB[k], (short)0, h1[m], false, false);
    }
  }

  // Store H transposed: Htr[c][j] so the vertical B fragment is a contiguous v2f.
#pragma unroll
  for (int m = 0; m < 5; ++m) {
    float* hm = hts + m * HT_FLOATS;
    const int rb = 8 * hi;                 // D rows: v + 8*hi
#pragma unroll
    for (int v = 0; v < 8; ++v) hm[lo * HS + rb + v] = h0[m][v];
    if (hi == 0) {                         // D1 lanes 0-15 carry rows 16..23
#pragma unroll
      for (int v = 0; v < 8; ++v) hm[lo * HS + 16 + v] = h1[m][v];
    }
  }

  // ---- Vertical pass: U(16x16) = Wv(16x24) x H(24x16).
  v8f u[5];
#pragma unroll
  for (int m = 0; m < 5; ++m) u[m] = vzero8();

#pragma unroll
  for (int k = 0; k < 6; ++k) {
    const int j0 = 4 * k + 2 * hi;         // B-fragment K rows (j0, j0+1)
#pragma unroll
    for (int m = 0; m < 5; ++m) {
      const float* hm = hts + m * HT_FLOATS;
      const v2f B = *(const v2f*)(hm + lo * HS + j0);
      u[m] = __builtin_amdgcn_wmma_f32_16x16x4_f32(
          false, wB[k], false, B, (short)0, u[m], false, false);
    }
  }

  // ---- SSIM elementwise + edge mask + reduction.
  const float C1 = 0.01f * 0.01f;
  const float C2 = 0.03f * 0.03f;
  const float cov_norm = 49.0f / 48.0f;
  float local = 0.0f;
#pragma unroll
  for (int v = 0; v < 8; ++v) {
    const int orow = r0 + v + 8 * hi;      // D layout: M = v + 8*(lane>=16)
    const int ocol = c0 + lo;
    const float ux = u[0][v], uy = u[1][v];
    const float uxx = u[2][v], uyy = u[3][v], uxy = u[4][v];
    const float vx  = cov_norm * (uxx - ux * ux);
    const float vy  = cov_norm * (uyy - uy * uy);
    const float vxy = cov_norm * (uxy - ux * uy);
    const float S = ((2.0f * ux * uy + C1) * (2.0f * vxy + C2)) /
                    ((ux * ux + uy * uy + C1) * (vx + vy + C2));
    if (orow < 506 && ocol < 506) local += S;
  }

  // wave32 butterfly reduction
#pragma unroll
  for (int off = 16; off > 0; off >>= 1)
    local += __shfl_xor(local, off, 32);

  if (lane == 0) {
    const float invN = 1.0f / (96.0f * 506.0f * 506.0f);
    atomicAdd(out, local * invN);
  }
}

__global__ void zero_out_kernel(float* o) { o[0] = 0.0f; }

extern "C" void kernel_launch(void* const* d_in, const int* in_sizes, int n_in,
                              void* d_out, int out_size, void* d_ws, size_t ws_size,
                              hipStream_t stream) {
  (void)in_sizes; (void)n_in; (void)out_size; (void)d_ws; (void)ws_size;
  const float* X = (const float*)d_in[0];   // input_tensor: 32*3*512*512 f32
  const float* Y = (const float*)d_in[1];   // target
  float* out = (float*)d_out;               // scalar mean

  zero_out_kernel<<<1, 1, 0, stream>>>(out);

  const int tiles = 96 * 32 * 32;           // 98304 tiles, 1 wave each
  ssim_wmma_kernel<<<tiles / WAVES, 32 * WAVES, 0, stream>>>(X, Y, out);
}